// TimeAwareMultiHeadAttention_60249801228416
// MI455X (gfx1250) — compile-verified
//
#include <hip/hip_runtime.h>
#include <hip/hip_bf16.h>

// CDNA5 / gfx1250, wave32. f32 WMMA (V_WMMA_F32_16X16X4_F32) for GEMM-shaped work.
typedef __attribute__((ext_vector_type(2))) float v2f;
typedef __attribute__((ext_vector_type(8))) float v8f;

constexpr int Bn = 32, Ln = 200, Hn = 128, NHn = 4, HDn = 32;
constexpr float SCALE = 0.17677669529663687f;   // 1/sqrt(32)
constexpr float NEGV  = -4294967295.0f;         // -2^32 + 1 (f32-rounded)

// ---------------------------------------------------------------------------
// Kernel 1: fused QKV projection.  Out = X @ W^T + b  for three weight sets.
// One wave per 16x16 output tile, K-loop over H=128 in steps of 4 (32 WMMAs).
// A layout (16x4 f32): lane l -> m=l%16, holds K = 2*(l/16)+{0,1} in v2f.
// B layout (4x16 f32): lane l -> n=l%16, holds K = 2*(l/16)+{0,1} in v2f.
// C/D layout: lane l, reg v -> (m = v + 8*(l/16), n = l%16).
// ---------------------------------------------------------------------------
__global__ __launch_bounds__(32)
void qkv_proj_kernel(const float* __restrict__ queries, const float* __restrict__ keys,
                     const float* __restrict__ Wq, const float* __restrict__ bq,
                     const float* __restrict__ Wk, const float* __restrict__ bk,
                     const float* __restrict__ Wv, const float* __restrict__ bv,
                     float* __restrict__ Qp, float* __restrict__ Kp, float* __restrict__ Vp)
{
    const int lane = threadIdx.x;
    const int g  = lane >> 4;       // 0/1: lane half
    const int ln = lane & 15;
    const int row0 = blockIdx.x * 16;          // 0..6384 (exactly 400 tiles)
    const int ct   = blockIdx.y;               // 0..23
    const int sel  = ct >> 3;                  // 0=Q, 1=K, 2=V
    const int n0   = (ct & 7) * 16;

    const float* X    = (sel == 0) ? queries : keys;
    const float* W    = (sel == 0) ? Wq : (sel == 1 ? Wk : Wv);
    const float* bias = (sel == 0) ? bq : (sel == 1 ? bk : bv);
    float*       D    = (sel == 0) ? Qp : (sel == 1 ? Kp : Vp);

    // bias is per output column n -> same for every m (every C reg)
    v8f c;
    const float bn = bias[n0 + ln];
#pragma unroll
    for (int v = 0; v < 8; ++v) c[v] = bn;

    const float* xrow = X + (size_t)(row0 + ln) * Hn;   // A row m = ln
    const float* wrow = W + (size_t)(n0   + ln) * Hn;   // B col n = ln (W^T -> contiguous k)

#pragma unroll
    for (int kk = 0; kk < Hn; kk += 4) {
        v2f a = *(const v2f*)(xrow + kk + 2 * g);
        v2f b = *(const v2f*)(wrow + kk + 2 * g);
        c = __builtin_amdgcn_wmma_f32_16x16x4_f32(false, a, false, b, (short)0, c,
                                                  false, false);
    }
#pragma unroll
    for (int v = 0; v < 8; ++v) {
        const int m = v + 8 * g;
        D[(size_t)(row0 + m) * Hn + n0 + ln] = c[v];
    }
}

// ---------------------------------------------------------------------------
// Kernel 2: attention.  One wave per (b, head, 16-query tile).
//   Phase A: S[16][200] = (Q K^T + Q.tK) * scale, masked -> LDS (WMMA + VALU dot)
//   Phase B: row softmax in LDS (exact reference semantics incl. all-masked rows)
//   Phase C: O = P V (WMMA) + P.tV (VALU, streaming), normalize, store.
// ---------------------------------------------------------------------------
__global__ __launch_bounds__(32)
void attn_kernel(const float* __restrict__ Qp, const float* __restrict__ Kp,
                 const float* __restrict__ Vp,
                 const float* __restrict__ tK, const float* __restrict__ tV,
                 const int* __restrict__ tmask, float* __restrict__ out)
{
    __shared__ float S[16][208];      // scores -> probabilities (13 k-tiles, padded)
    __shared__ float Qs[16][32];      // Q tile rows
    __shared__ float Mb[16][32];      // merge buffer for WMMA O
    __shared__ float linv[16];        // 1 / softmax denominator

    const int lane = threadIdx.x;
    const int g  = lane >> 4;
    const int ln = lane & 15;
    const int bh = blockIdx.x;
    const int b  = bh / NHn;
    const int h  = bh % NHn;
    const int q0 = blockIdx.y * 16;
    const int hd0 = h * HDn;

    // ---- load Q tile into LDS (coalesced: 32 lanes over HD) ----
    for (int m = 0; m < 16; ++m) {
        int qq = q0 + m;
        int qc = qq < Ln ? qq : Ln - 1;
        Qs[m][lane] = Qp[(size_t)(b * Ln + qc) * Hn + hd0 + lane];
    }
    __syncthreads();

    // ---- A-operand registers for S = Q K^T (8 chunks of K=4 over HD=32) ----
    v2f aq[8];
#pragma unroll
    for (int j = 0; j < 8; ++j)
        aq[j] = *(const v2f*)(&Qs[ln][4 * j + 2 * g]);

    // per-C-reg row predicates (query row out of range or time-masked)
    bool rowbad[8];
#pragma unroll
    for (int v = 0; v < 8; ++v) {
        const int qq = q0 + v + 8 * g;
        rowbad[v] = (qq >= Ln) ? true : (tmask[b * Ln + qq] != 0);
    }

    // =================== Phase A: scores ===================
    for (int kt = 0; kt < 13; ++kt) {
        const int k0 = kt * 16;
        float sv[8];
        if (k0 <= q0 + 15) {                       // tile touches causal region
            v8f c = {};
            const int kcol = k0 + ln;
            const int kc   = kcol < Ln ? kcol : Ln - 1;
            const float* krow = Kp + (size_t)(b * Ln + kc) * Hn + hd0;
#pragma unroll
            for (int j = 0; j < 8; ++j) {
                v2f bb = *(const v2f*)(krow + 4 * j + 2 * g);
                c = __builtin_amdgcn_wmma_f32_16x16x4_f32(false, aq[j], false, bb,
                                                          (short)0, c, false, false);
            }
#pragma unroll
            for (int v = 0; v < 8; ++v) {
                const int m  = v + 8 * g;
                const int qq = q0 + m;
                const int kk = k0 + ln;
                const bool masked = rowbad[v] || (kk >= Ln) || (kk > qq);
                if (!masked) {
                    // additive relative-time bias: dot(tK[b,q,k,hd0:hd0+32], Q[q,:])
                    const float4* tkp = (const float4*)(tK +
                        ((size_t)(b * Ln + qq) * Ln + kk) * Hn + hd0);
                    const float4* qrp = (const float4*)(&Qs[m][0]);
                    float acc = 0.f;
#pragma unroll
                    for (int i = 0; i < 8; ++i) {
                        float4 t = tkp[i], q = qrp[i];
                        acc += t.x * q.x + t.y * q.y + t.z * q.z + t.w * q.w;
                    }
                    sv[v] = (c[v] + acc) * SCALE;
                } else {
                    sv[v] = NEGV;                  // reference masks AFTER scaling
                }
            }
        } else {                                   // fully above diagonal: no loads
#pragma unroll
            for (int v = 0; v < 8; ++v) sv[v] = NEGV;
        }
#pragma unroll
        for (int v = 0; v < 8; ++v) S[v + 8 * g][k0 + ln] = sv[v];
    }
    __syncthreads();

    // =================== Phase B: softmax (lane r owns row r) ===================
    if (lane < 16) {
        const int r = lane;
        float mx = -3.4028235e38f;
        for (int k = 0; k < Ln; ++k) mx = fmaxf(mx, S[r][k]);
        float sum = 0.f;
        for (int k = 0; k < Ln; ++k) {
            float p = __expf(S[r][k] - mx);        // all-masked row -> p==1 everywhere
            S[r][k] = p;
            sum += p;
        }
        for (int k = Ln; k < 208; ++k) S[r][k] = 0.f;  // kill pad columns for WMMA
        linv[r] = 1.0f / sum;                      // uniform 1/200 matches reference
    }
    __syncthreads();

    // =================== Phase C: output ===================
    v8f oc0 = {}, oc1 = {};                        // WMMA accum: d = 0..15 / 16..31
    float ot[16];                                  // tV accum, lane layout:
#pragma unroll                                     //   row r = lane%16, d-half = lane/16
    for (int d = 0; d < 16; ++d) ot[d] = 0.f;

    const int r  = ln;
    const int h2 = g;
    const int qrc = (q0 + r) < Ln ? (q0 + r) : Ln - 1;
    const float* tvbase = tV + ((size_t)(b * Ln + qrc) * Ln) * Hn + hd0 + 16 * h2;

    for (int kt = 0; kt < 13; ++kt) {
        const int k0 = kt * 16;
        if (kt + 1 < 13)
            __builtin_prefetch(tvbase + (size_t)(k0 + 16) * Hn, 0, 0);

        // ---- WMMA: O += P @ V  (A from LDS, B streamed from Vp) ----
#pragma unroll
        for (int j = 0; j < 4; ++j) {
            const int kA = k0 + 4 * j + 2 * g;
            v2f a = *(const v2f*)(&S[ln][kA]);     // P in A layout (m = ln)
            const int kr0 = kA     < Ln ? kA     : Ln - 1;  // P==0 past L -> safe
            const int kr1 = kA + 1 < Ln ? kA + 1 : Ln - 1;
            const float* v0 = Vp + (size_t)(b * Ln + kr0) * Hn + hd0;
            const float* v1 = Vp + (size_t)(b * Ln + kr1) * Hn + hd0;
            v2f b0, b1;
            b0.x = v0[ln];      b0.y = v1[ln];
            b1.x = v0[16 + ln]; b1.y = v1[16 + ln];
            oc0 = __builtin_amdgcn_wmma_f32_16x16x4_f32(false, a, false, b0, (short)0,
                                                        oc0, false, false);
            oc1 = __builtin_amdgcn_wmma_f32_16x16x4_f32(false, a, false, b1, (short)0,
                                                        oc1, false, false);
        }

        // ---- VALU: ot += p * tV[b,q,k,:]  (the per-query value bias) ----
        for (int k = 0; k < 16; ++k) {
            const int kk = k0 + k;
            const float p = S[r][kk];
            if (__any((int)(p != 0.0f))) {         // wave-uniform skip of dead tiles
                const int kkc = kk < Ln ? kk : Ln - 1;
                const float4* tvp = (const float4*)(tvbase + (size_t)kkc * Hn);
#pragma unroll
                for (int i = 0; i < 4; ++i) {
                    float4 t = tvp[i];
                    ot[4 * i + 0] += p * t.x;
                    ot[4 * i + 1] += p * t.y;
                    ot[4 * i + 2] += p * t.z;
                    ot[4 * i + 3] += p * t.w;
                }
            }
        }
    }

    // ---- merge WMMA layout with row layout via LDS, normalize, store ----
#pragma unroll
    for (int v = 0; v < 8; ++v) {
        const int m = v + 8 * g;
        Mb[m][ln]      = oc0[v];
        Mb[m][16 + ln] = oc1[v];
    }
    __syncthreads();

    if (q0 + r < Ln) {
        const float li = linv[r];
        float* op = out + (size_t)(b * Ln + q0 + r) * Hn + hd0 + 16 * h2;
#pragma unroll
        for (int d = 0; d < 16; ++d)
            op[d] = (Mb[r][16 * h2 + d] + ot[d]) * li;
    }
}

// ---------------------------------------------------------------------------
extern "C" void kernel_launch(void* const* d_in, const int* in_sizes, int n_in,
                              void* d_out, int out_size, void* d_ws, size_t ws_size,
                              hipStream_t stream)
{
    const float* queries = (const float*)d_in[0];
    const float* keys    = (const float*)d_in[1];
    const float* tK      = (const float*)d_in[2];
    const float* tV      = (const float*)d_in[3];
    const float* Wq      = (const float*)d_in[4];
    const float* bq      = (const float*)d_in[5];
    const float* Wk      = (const float*)d_in[6];
    const float* bk      = (const float*)d_in[7];
    const float* Wv      = (const float*)d_in[8];
    const float* bv      = (const float*)d_in[9];
    const int*   tmask   = (const int*)d_in[10];
    // d_in[11] = attn_mask: causal triu(k=1) -> computed analytically in-kernel.

    float* outp = (float*)d_out;
    float* Qp = (float*)d_ws;                       // 3 x 6400x128 f32 = 9.8 MB scratch
    float* Kp = Qp + (size_t)Bn * Ln * Hn;
    float* Vp = Kp + (size_t)Bn * Ln * Hn;

    dim3 g1(Bn * Ln / 16, 24);                      // 400 row tiles x (3 mats x 8 col tiles)
    qkv_proj_kernel<<<g1, 32, 0, stream>>>(queries, keys, Wq, bq, Wk, bk, Wv, bv,
                                           Qp, Kp, Vp);

    dim3 g2(Bn * NHn, (Ln + 15) / 16);              // 128 x 13 waves
    attn_kernel<<<g2, 32, 0, stream>>>(Qp, Kp, Vp, tK, tV, tmask, outp);
}